// LightGCNModel_21638045237421
// MI455X (gfx1250) — compile-verified
//
#include <hip/hip_runtime.h>
#include <hip/hip_bf16.h>

#define DIM 32

typedef __attribute__((ext_vector_type(2))) float v2f;
typedef __attribute__((ext_vector_type(8))) float v8f;

// ---------------------------------------------------------------------------
// SpMM scatter:  dst[row] += val * src[col]   (segment_sum over rows)
// 8 lanes per edge, each lane handles 4 contiguous dims:
//   - 128-bit coalesced gather of the source row slice
//   - 4 contiguous native f32 atomics into the destination row slice (L2-resident)
// Layer 1 reads the virtual concat(user_emb, item_emb) directly (src == nullptr).
// ---------------------------------------------------------------------------
__global__ void lightgcn_scatter(const float* __restrict__ ue,
                                 const float* __restrict__ ie,
                                 const float* __restrict__ src,
                                 float* __restrict__ dst,
                                 const float* __restrict__ vals,
                                 const int* __restrict__ rows,
                                 const int* __restrict__ cols,
                                 long long n8, int num_users) {
  long long t = (long long)blockIdx.x * blockDim.x + threadIdx.x;
  if (t >= n8) return;
  long long e = t >> 3;          // edge id
  int q = (int)(t & 7);          // which 4-dim eighth of the 32-dim row
  int r = rows[e];
  int c = cols[e];
  float v = vals[e];

  const float* s;
  if (src) {
    s = src + (size_t)c * DIM;
  } else {
    s = (c < num_users) ? (ue + (size_t)c * DIM)
                        : (ie + (size_t)(c - num_users) * DIM);
  }
  float4 x = *(const float4*)(s + q * 4);
  float* d = dst + (size_t)r * DIM + q * 4;
  unsafeAtomicAdd(d + 0, v * x.x);   // global_atomic_add_f32
  unsafeAtomicAdd(d + 1, v * x.y);
  unsafeAtomicAdd(d + 2, v * x.z);
  unsafeAtomicAdd(d + 3, v * x.w);
}

// ---------------------------------------------------------------------------
// Scoring via V_WMMA_F32_16X16X4_F32.
// One wave handles a tile of 16 batch elements.
//   A (16x32) row m   = final[user_idx[b]]          (= (e0+e1+e2) row, scaled later)
//   B (32x16) col n   = final[pos] - final[neg]
//   D = A x B ; diagonal D[m][m] = pos_score - neg_score (x 1/9 for the /3 /3)
// A f32 16x4 lane layout (ISA 7.12.2): lanes 0-15 hold K={4k,4k+1}, lanes 16-31
// hold K={4k+2,4k+3}; B mirrors it. A consistent K-permutation between A and B
// leaves the inner product invariant, so the reduction is exact regardless.
// D layout: diag[m] -> lane m, vgpr m (m<8); lane m+16, vgpr m-8 (m>=8).
// ---------------------------------------------------------------------------
__global__ void lightgcn_score(const float* __restrict__ ue,
                               const float* __restrict__ ie,
                               const float* __restrict__ e1,
                               const float* __restrict__ e2,
                               const int* __restrict__ uidx,
                               const int* __restrict__ pidx,
                               const int* __restrict__ nidx,
                               float* __restrict__ out,
                               int num_users, int batch) {
  int wave = (int)((blockIdx.x * (long long)blockDim.x + threadIdx.x) >> 5);
  int lane = threadIdx.x & 31;
  int m    = lane & 15;
  int half = lane >> 4;

  int b = wave * 16 + m;
  if (b >= batch) b = batch - 1;           // clamp loads; stores guarded below
  int ur = uidx[b];
  int pr = pidx[b];
  int nr = nidx[b];

  const float* u0 = ue + (size_t)ur * DIM;
  const float* u1 = e1 + (size_t)ur * DIM;
  const float* u2 = e2 + (size_t)ur * DIM;
  const float* p0 = ie + (size_t)pr * DIM;
  const float* p1 = e1 + (size_t)(num_users + pr) * DIM;
  const float* p2 = e2 + (size_t)(num_users + pr) * DIM;
  const float* n0 = ie + (size_t)nr * DIM;
  const float* n1 = e1 + (size_t)(num_users + nr) * DIM;
  const float* n2 = e2 + (size_t)(num_users + nr) * DIM;

  v8f acc = {};
#pragma unroll
  for (int k = 0; k < 8; ++k) {
    int d0 = 4 * k + 2 * half;
    v2f A, B;
    A.x = u0[d0]     + u1[d0]     + u2[d0];
    A.y = u0[d0 + 1] + u1[d0 + 1] + u2[d0 + 1];
    float pv0 = p0[d0]     + p1[d0]     + p2[d0];
    float pv1 = p0[d0 + 1] + p1[d0 + 1] + p2[d0 + 1];
    float nv0 = n0[d0]     + n1[d0]     + n2[d0];
    float nv1 = n0[d0 + 1] + n1[d0 + 1] + n2[d0 + 1];
    B.x = pv0 - nv0;
    B.y = pv1 - nv1;
    acc = __builtin_amdgcn_wmma_f32_16x16x4_f32(
        /*neg_a=*/false, A, /*neg_b=*/false, B,
        /*c_mod=*/(short)0, acc, /*reuse_a=*/false, /*reuse_b=*/false);
  }

  // Extract diagonal element owned by this lane (if any).
  int sel  = (lane < 8) ? lane : ((lane >= 24) ? (lane - 24) : 0);
  float dv;
  switch (sel) {
    case 0: dv = acc[0]; break;
    case 1: dv = acc[1]; break;
    case 2: dv = acc[2]; break;
    case 3: dv = acc[3]; break;
    case 4: dv = acc[4]; break;
    case 5: dv = acc[5]; break;
    case 6: dv = acc[6]; break;
    default: dv = acc[7]; break;
  }
  int outm = (lane < 8) ? lane : ((lane >= 24) ? (lane - 16) : -1);
  if (outm >= 0) {
    int ob = wave * 16 + outm;
    if (ob < batch) out[ob] = dv * (1.0f / 9.0f);   // (/3 user) * (/3 item)
  }
}

extern "C" void kernel_launch(void* const* d_in, const int* in_sizes, int n_in,
                              void* d_out, int out_size, void* d_ws, size_t ws_size,
                              hipStream_t stream) {
  const float* ue   = (const float*)d_in[0];
  const float* ie   = (const float*)d_in[1];
  const float* vals = (const float*)d_in[2];
  const int*   rows = (const int*)d_in[3];
  const int*   cols = (const int*)d_in[4];
  const int*   uidx = (const int*)d_in[5];
  const int*   pidx = (const int*)d_in[6];
  const int*   nidx = (const int*)d_in[7];

  const int       num_users = in_sizes[0] / DIM;
  const int       num_items = in_sizes[1] / DIM;
  const long long nnz       = in_sizes[2];
  const int       batch     = in_sizes[5];
  const long long N         = (long long)num_users + num_items;
  const size_t    emb_elems = (size_t)N * DIM;

  float* e1 = (float*)d_ws;             // S * e0
  float* e2 = e1 + emb_elems;           // S * S * e0  (total ws: 2*N*DIM*4 = 76.8 MB)

  const long long n8 = nnz * 8;
  const int threads = 256;
  const unsigned scat_blocks = (unsigned)((n8 + threads - 1) / threads);

  // Layer 1: e1 = S * concat(ue, ie)
  hipMemsetAsync(e1, 0, emb_elems * sizeof(float), stream);
  lightgcn_scatter<<<scat_blocks, threads, 0, stream>>>(
      ue, ie, /*src=*/nullptr, e1, vals, rows, cols, n8, num_users);

  // Layer 2: e2 = S * e1
  hipMemsetAsync(e2, 0, emb_elems * sizeof(float), stream);
  lightgcn_scatter<<<scat_blocks, threads, 0, stream>>>(
      ue, ie, /*src=*/e1, e2, vals, rows, cols, n8, num_users);

  // Scoring: one wave per 16 batch rows, WMMA f32 16x16x4 accumulation.
  const int tiles = (batch + 15) / 16;
  const int waves_per_block = threads / 32;
  const unsigned score_blocks = (unsigned)((tiles + waves_per_block - 1) / waves_per_block);
  lightgcn_score<<<score_blocks, threads, 0, stream>>>(
      ue, ie, e1, e2, uidx, pidx, nidx, (float*)d_out, num_users, batch);
}